// Embedded_GCN_78915729096976
// MI455X (gfx1250) — compile-verified
//
#include <hip/hip_runtime.h>
#include <hip/hip_bf16.h>

typedef __attribute__((ext_vector_type(2))) float v2f;
typedef __attribute__((ext_vector_type(8))) float v8f;
typedef __attribute__((ext_vector_type(4))) unsigned int u32x4;
typedef __attribute__((ext_vector_type(8))) int i32x8;
typedef __attribute__((ext_vector_type(4))) int i32x4;

// Problem dims (fixed by the reference harness)
#define NN   4096   // nodes
#define BB   32     // batch
#define CI   32     // C_in
#define CO   64     // C_out
#define DD   16     // embed dim
#define JJ   (BB*CI)   // 1024 flattened (b,c) columns

// ---------------------------------------------------------------------------
// 1) A[n,m] = softmax_m( relu( E[n]·E[m] ) )
// ---------------------------------------------------------------------------
__global__ __launch_bounds__(256) void adj_softmax(const float* __restrict__ E,
                                                   float* __restrict__ A) {
    const int n = blockIdx.x;
    const int t = threadIdx.x;
    __shared__ float sE[DD];
    __shared__ float sc[NN];
    __shared__ float red[256];

    if (t < DD) sE[t] = E[n * DD + t];
    __syncthreads();

    float lmax = -1e30f;
    for (int m = t; m < NN; m += 256) {
        const float4* em = reinterpret_cast<const float4*>(E + (size_t)m * DD);
        float4 e0 = em[0], e1 = em[1], e2 = em[2], e3 = em[3];
        float d = e0.x*sE[0]  + e0.y*sE[1]  + e0.z*sE[2]  + e0.w*sE[3]
                + e1.x*sE[4]  + e1.y*sE[5]  + e1.z*sE[6]  + e1.w*sE[7]
                + e2.x*sE[8]  + e2.y*sE[9]  + e2.z*sE[10] + e2.w*sE[11]
                + e3.x*sE[12] + e3.y*sE[13] + e3.z*sE[14] + e3.w*sE[15];
        d = fmaxf(d, 0.0f);                    // relu
        sc[m] = d;
        lmax = fmaxf(lmax, d);
    }
    red[t] = lmax;
    __syncthreads();
    for (int s = 128; s > 0; s >>= 1) {
        if (t < s) red[t] = fmaxf(red[t], red[t + s]);
        __syncthreads();
    }
    const float gmax = red[0];
    __syncthreads();

    float lsum = 0.0f;
    for (int m = t; m < NN; m += 256) {
        float e = __expf(sc[m] - gmax);
        sc[m] = e;
        lsum += e;
    }
    red[t] = lsum;
    __syncthreads();
    for (int s = 128; s > 0; s >>= 1) {
        if (t < s) red[t] += red[t + s];
        __syncthreads();
    }
    const float inv = 1.0f / red[0];
    for (int m = t; m < NN; m += 256)
        A[(size_t)n * NN + m] = sc[m] * inv;
}

// ---------------------------------------------------------------------------
// 2) Xt[m, b*CI+c] = x[b, m, c]   ([B,N,C] -> [N, B*C])
// ---------------------------------------------------------------------------
__global__ __launch_bounds__(256) void transpose_x(const float* __restrict__ x,
                                                   float* __restrict__ Xt) {
    const int idx = blockIdx.x * 256 + threadIdx.x;    // 0 .. N*J-1
    const int m = idx >> 10;            // / 1024
    const int j = idx & 1023;
    const int b = j >> 5, c = j & 31;
    Xt[idx] = x[((size_t)b * NN + m) * CI + c];
}

// ---------------------------------------------------------------------------
// TDM: issue a 2D tile load Global -> LDS via descriptor (CDNA5 Tensor Data
// Mover, CDNA5 ISA). Padding inserts the bank-conflict-free
// LDS row stride. Tracked by TENSORcnt.
//   tile_c   : tile dim0 (contiguous elements per row)
//   tile_r   : tile dim1 (rows)
//   stride   : tensor dim0 stride (elements between rows)
//   pi, pa   : pad_interval code (2^(pi+1) DWORDs), pad_amount code (pa+1 DWORDs)
// ---------------------------------------------------------------------------
__device__ __forceinline__ void tdm_load_2d(const float* gptr, unsigned lds_off,
                                            unsigned tile_c, unsigned tile_r,
                                            unsigned tensor_c, unsigned tensor_r,
                                            unsigned stride,
                                            unsigned pi, unsigned pa) {
    const unsigned long long ga = (unsigned long long)(uintptr_t)gptr;
    u32x4 g0;
    g0.x = 1u;                                        // count=1, user descriptor
    g0.y = lds_off;                                   // lds_addr (bytes)
    g0.z = (unsigned)(ga & 0xFFFFFFFFull);            // global_addr[31:0]
    g0.w = (unsigned)((ga >> 32) & 0x01FFFFFFull)     // global_addr[56:32]
         | (2u << 30);                                // type=2 ("image")
    i32x8 g1;
    g1[0] = (int)((2u << 16)        // data_size = 4B
                | (1u << 20)        // pad_enable
                | (pi << 22) | (pa << 25));
    g1[1] = (int)(tensor_c << 16);                    // abar=0 | tensor_dim0 lo16
    g1[2] = (int)((tensor_c >> 16) | (tensor_r << 16));
    g1[3] = (int)((tensor_r >> 16) | (tile_c << 16)); // | tile_dim0
    g1[4] = (int)tile_r;                              // tile_dim1 (tile_dim2=0)
    g1[5] = (int)stride;                              // tensor_dim0_stride lo32
    g1[6] = 0;                                        // stride hi | dim1_stride lo
    g1[7] = 0;
    const i32x4 z4 = {0, 0, 0, 0};
    const i32x8 z8 = {0, 0, 0, 0, 0, 0, 0, 0};
    // amdgpu-toolchain (clang-23) 6-arg form
    __builtin_amdgcn_tensor_load_to_lds(g0, g1, z4, z4, z8, 0);
}

// ---------------------------------------------------------------------------
// 3) Cout = alpha * (A @ B) + beta * Cin     A:[4096,4096]  B,Cin,Cout:[4096,1024]
//    Block tile 128x64, 8 waves, each wave = 32x32 (2x2 WMMA f32 16x16x4 tiles).
//    Tiles staged by TDM into double-buffered padded LDS; wave 0 drives the DMA.
// ---------------------------------------------------------------------------
#define TM 128
#define TN 64
#define TK 32
#define LDA 36   // 32 + pad 4  (TDM pad: 4 DWORDs every 32  -> codes pi=4, pa=3)
#define LDB 80   // 64 + pad 16 (TDM pad: 16 DWORDs every 64 -> codes pi=5, pa=15)

__global__ __launch_bounds__(256) void gemm_f32(const float* __restrict__ Amat,
                                                const float* __restrict__ Bmat,
                                                const float* __restrict__ Cin,
                                                float* __restrict__ Cout,
                                                float alpha, float beta) {
    __shared__ float sA[2][TM * LDA];
    __shared__ float sB[2][TK * LDB];

    const int t  = threadIdx.x;
    const int l  = t & 31, w = t >> 5;
    const int hl = l >> 4, lr = l & 15;
    const int wm = w & 3,  wn = w >> 2;          // 4x2 wave grid over 128x64
    const int rblk = blockIdx.y * TM;
    const int cblk = blockIdx.x * TN;
    const bool leader = (w == 0);                // wave 0 drives the TDM

    v8f acc[2][2] = {{{}, {}}, {{}, {}}};

    if (leader) {
        tdm_load_2d(Amat + (size_t)rblk * NN,
                    (unsigned)(uintptr_t)&sA[0][0], TK, TM, NN, NN, NN, 4, 3);
        tdm_load_2d(Bmat + cblk,
                    (unsigned)(uintptr_t)&sB[0][0], TN, TK, JJ, NN, JJ, 5, 15);
        __builtin_amdgcn_s_wait_tensorcnt(0);
    }
    __syncthreads();

    int buf = 0;
    for (int kb = 0; kb < NN; kb += TK) {
        // Prefetch next K-tile into the other buffer while we compute.
        if (leader && (kb + TK < NN)) {
            tdm_load_2d(Amat + (size_t)rblk * NN + kb + TK,
                        (unsigned)(uintptr_t)&sA[buf ^ 1][0],
                        TK, TM, NN, NN, NN, 4, 3);
            tdm_load_2d(Bmat + (size_t)(kb + TK) * JJ + cblk,
                        (unsigned)(uintptr_t)&sB[buf ^ 1][0],
                        TN, TK, JJ, NN, JJ, 5, 15);
        }

        #pragma unroll
        for (int k0 = 0; k0 < TK; k0 += 4) {
            v2f af[2], bf[2];
            #pragma unroll
            for (int i = 0; i < 2; ++i) {
                const int row = wm * 32 + i * 16 + lr;
                af[i].x = sA[buf][row * LDA + k0 + 2 * hl];
                af[i].y = sA[buf][row * LDA + k0 + 2 * hl + 1];
            }
            #pragma unroll
            for (int j = 0; j < 2; ++j) {
                const int col = wn * 32 + j * 16 + lr;
                bf[j].x = sB[buf][(k0 + 2 * hl)     * LDB + col];
                bf[j].y = sB[buf][(k0 + 2 * hl + 1) * LDB + col];
            }
            #pragma unroll
            for (int i = 0; i < 2; ++i)
                #pragma unroll
                for (int j = 0; j < 2; ++j)
                    acc[i][j] = __builtin_amdgcn_wmma_f32_16x16x4_f32(
                        false, af[i], false, bf[j], (short)0, acc[i][j],
                        false, false);
        }

        if (leader) __builtin_amdgcn_s_wait_tensorcnt(0);
        __syncthreads();   // everyone done with buf; next tile landed in buf^1
        buf ^= 1;
    }

    #pragma unroll
    for (int i = 0; i < 2; ++i)
        #pragma unroll
        for (int j = 0; j < 2; ++j) {
            const int gn = cblk + wn * 32 + j * 16 + lr;
            #pragma unroll
            for (int v = 0; v < 8; ++v) {
                const int gm = rblk + wm * 32 + i * 16 + v + 8 * hl;
                const size_t idx = (size_t)gm * JJ + gn;
                float val = alpha * acc[i][j][v];
                if (beta != 0.0f) val += beta * Cin[idx];
                Cout[idx] = val;
            }
        }
}

// ---------------------------------------------------------------------------
// 4) Per-node output: W[n] = sum_d E[n,d] wp[d]; out[b,n,:] = x_g[b,n,:]@W[n] + bias[n]
//    8 waves, each one 16x16 WMMA tile of the 32x64 per-node GEMM (K=96)
// ---------------------------------------------------------------------------
#define LWN 80    // padded stride for W tile in LDS
#define LXG 100   // padded stride for x_g tile in LDS

__global__ __launch_bounds__(256) void node_out(const float* __restrict__ E,
                                                const float* __restrict__ wp,
                                                const float* __restrict__ bp,
                                                const float* __restrict__ Xt,
                                                const float* __restrict__ Y1,
                                                const float* __restrict__ Y2,
                                                float* __restrict__ out) {
    const int n = blockIdx.x;
    const int t = threadIdx.x;
    __shared__ float sE[DD];
    __shared__ float sWn[96 * LWN];   // [ki][o]
    __shared__ float sXg[BB * LXG];   // [b][ki]
    __shared__ float sBias[CO];

    if (t < DD) sE[t] = E[n * DD + t];
    __syncthreads();

    // Generate per-node weights: W[ki,o] = sum_d E[n,d] * wp[d,ki,o]
    for (int e = t; e < 96 * CO; e += 256) {
        const int ki = e >> 6, o = e & 63;
        float wv = 0.0f;
        #pragma unroll
        for (int d = 0; d < DD; ++d)
            wv += sE[d] * wp[(size_t)d * (96 * CO) + e];
        sWn[ki * LWN + o] = wv;
    }
    if (t < CO) {
        float bv = 0.0f;
        #pragma unroll
        for (int d = 0; d < DD; ++d) bv += sE[d] * bp[d * CO + t];
        sBias[t] = bv;
    }
    // Gather x_g[b, ki] : k=0 -> X, k=1 -> A@X, k=2 -> 2A(A@X) - X
    for (int e = t; e < BB * 96; e += 256) {
        const int b = e / 96, ki = e % 96;
        const int k = ki >> 5, i = ki & 31;
        const float* src = (k == 0) ? Xt : (k == 1 ? Y1 : Y2);
        sXg[b * LXG + ki] = src[(size_t)n * JJ + (b << 5) + i];
    }
    __syncthreads();

    const int l  = t & 31, w = t >> 5;
    const int hl = l >> 4, lr = l & 15;
    const int mt = w & 1, nt = w >> 1;   // 2x4 wave grid over 32x64 output

    v8f acc = {};
    #pragma unroll
    for (int k0 = 0; k0 < 96; k0 += 4) {
        v2f af, bf;
        const int row = mt * 16 + lr;
        af.x = sXg[row * LXG + k0 + 2 * hl];
        af.y = sXg[row * LXG + k0 + 2 * hl + 1];
        const int col = nt * 16 + lr;
        bf.x = sWn[(k0 + 2 * hl)     * LWN + col];
        bf.y = sWn[(k0 + 2 * hl + 1) * LWN + col];
        acc = __builtin_amdgcn_wmma_f32_16x16x4_f32(
            false, af, false, bf, (short)0, acc, false, false);
    }
    #pragma unroll
    for (int v = 0; v < 8; ++v) {
        const int b = mt * 16 + v + 8 * hl;
        const int o = nt * 16 + lr;
        out[((size_t)b * NN + n) * CO + o] = acc[v] + sBias[o];
    }
}

// ---------------------------------------------------------------------------
extern "C" void kernel_launch(void* const* d_in, const int* in_sizes, int n_in,
                              void* d_out, int out_size, void* d_ws, size_t ws_size,
                              hipStream_t stream) {
    const float* x  = (const float*)d_in[0];   // [32,4096,32]
    const float* E  = (const float*)d_in[1];   // [4096,16]
    // d_in[2] = laplacian_mx (unused by forward)
    const float* wp = (const float*)d_in[3];   // [16,3,32,64]
    const float* bp = (const float*)d_in[4];   // [16,64]
    float* out = (float*)d_out;                // [32,4096,64]

    // Workspace layout (~117.5 MB)
    float* Aws = (float*)d_ws;                 // [4096,4096]
    float* Xt  = Aws + (size_t)NN * NN;        // [4096,1024]
    float* Y1  = Xt  + (size_t)NN * JJ;        // [4096,1024]
    float* Y2  = Y1  + (size_t)NN * JJ;        // [4096,1024]

    // 1) adaptive adjacency
    adj_softmax<<<NN, 256, 0, stream>>>(E, Aws);
    // 2) flatten x to [N, B*C]
    transpose_x<<<(NN * JJ) / 256, 256, 0, stream>>>(x, Xt);
    // 3) y1 = A @ X
    dim3 ggrid(JJ / TN, NN / TM);
    gemm_f32<<<ggrid, 256, 0, stream>>>(Aws, Xt, Xt, Y1, 1.0f, 0.0f);
    // 4) y2 = 2*A@y1 - X   (Chebyshev T2 applied to x, no NxN A@A needed)
    gemm_f32<<<ggrid, 256, 0, stream>>>(Aws, Y1, Xt, Y2, 2.0f, -1.0f);
    // 5) per-node grouped GEMM with generated weights + bias
    node_out<<<NN, 256, 0, stream>>>(E, wp, bp, Xt, Y1, Y2, out);
}